// DBScan_56822417326295
// MI455X (gfx1250) — compile-verified
//
#include <hip/hip_runtime.h>
#include <hip/hip_bf16.h>

// ---------------- problem constants (from reference) ----------------
#define NPTS     8192
#define NCOLS    10          // DIM(3) + 1 + 1 + NUM_CLASSES(5)
#define NTILE    512         // NPTS / 16
#define MIN_PTS  10
#define EPS2     (0.12f * 0.12f)
#define SENT     NPTS
#define OUTW     6           // DIM + 1 + 1 (data cols) + class col

typedef float v2f __attribute__((ext_vector_type(2)));
typedef float v8f __attribute__((ext_vector_type(8)));

__device__ __forceinline__ int imin(int a, int b) { return a < b ? a : b; }

// ---------------- kernel 0: per-point prep (padded coords, sq-norm, argmax class) ---
__global__ __launch_bounds__(256) void prep_kernel(const float* __restrict__ x,
                                                   float* __restrict__ coords4,
                                                   float* __restrict__ sq,
                                                   int* __restrict__ cls) {
    int i = blockIdx.x * blockDim.x + threadIdx.x;
    if (i >= NPTS) return;
    const float* row = x + (size_t)i * NCOLS;
    float c0 = row[0], c1 = row[1], c2 = row[2];
    coords4[(size_t)i * 4 + 0] = c0;     // K padded 3 -> 4 with a zero column so the
    coords4[(size_t)i * 4 + 1] = c1;     // WMMA fragments are plain b64 loads
    coords4[(size_t)i * 4 + 2] = c2;
    coords4[(size_t)i * 4 + 3] = 0.0f;
    sq[i] = c0 * c0 + c1 * c1 + c2 * c2;
    float best = row[5];
    int bi = 0;
#pragma unroll
    for (int c = 1; c < 5; ++c) {        // first-max semantics like jnp.argmax
        float v = row[5 + c];
        if (v > best) { best = v; bi = c; }
    }
    cls[i] = bi;
}

// ---------------- kernel 1: WMMA Gram tiles -> bit-packed adjacency + degrees -------
// One wave (32 threads) per 16-row strip. For each 16-col tile: one
// V_WMMA_F32_16X16X4_F32 gives 256 dot products; 8 ballots turn the eps/class
// tests into 16-bit row masks (rows v and v+8 are the two halves of ballot v).
// Packed layout: adjB[(rowtile*8 + v)*NTILE + jt] = ballot_v
//   -> low 16 bits: row rowtile*16+v, high 16 bits: row rowtile*16+v+8
__global__ __launch_bounds__(32) void adj_kernel(const float* __restrict__ coords4,
                                                 const float* __restrict__ sq,
                                                 const int* __restrict__ cls,
                                                 unsigned* __restrict__ adjB,
                                                 int* __restrict__ deg) {
    const int rt   = blockIdx.x;        // row tile 0..511
    const int lane = threadIdx.x;       // 0..31 (wave32)
    const int m    = lane & 15;
    const int kb   = (lane >> 4) * 2;   // 0 or 2 (K pair this lane holds)

    // A fragment: rows rt*16+m, K = kb..kb+1 (single unconditional b64 load)
    const v2f a = *(const v2f*)(coords4 + (size_t)(rt * 16 + m) * 4 + kb);

    // metadata for this lane's C/D rows: M = v + 8*(lane>>4)
    const int mg = (lane >> 4) * 8;
    float sr[8];
    int   cr[8];
#pragma unroll
    for (int v = 0; v < 8; ++v) {
        const int r = rt * 16 + mg + v;
        sr[v] = sq[r];
        cr[v] = cls[r];
    }

    unsigned* myRow = adjB + (size_t)(rt * 8 + lane) * NTILE; // valid when lane < 8
    unsigned degLo = 0, degHi = 0;      // lanes 0..7 own rows (lane) and (lane+8)

    for (int jt = 0; jt < NTILE; ++jt) {
        const int cj = jt * 16 + m;     // this lane's B point / C-column point
        const v2f b = *(const v2f*)(coords4 + (size_t)cj * 4 + kb);
        const float sqc = sq[cj];
        const int   clc = cls[cj];

        // prefetch the B-fragment strip a few tiles ahead (global_prefetch_b8)
        __builtin_prefetch(coords4 + (size_t)(((jt + 4) & (NTILE - 1)) * 16 + m) * 4, 0, 1);

        v8f c = {};
        v8f d = __builtin_amdgcn_wmma_f32_16x16x4_f32(
                    false, a, false, b, (short)0, c, false, false);

        unsigned bal[8];
#pragma unroll
        for (int v = 0; v < 8; ++v) {
            const float d2  = sr[v] + sqc - 2.0f * d[v];
            const bool pred = (d2 < EPS2) && (cr[v] == clc);
            bal[v] = __builtin_amdgcn_ballot_w32(pred);
        }
        // branch-free selection: lane v keeps ballot v (v compile-time -> cndmask chain)
        unsigned my = 0;
#pragma unroll
        for (int v = 0; v < 8; ++v)
            my = (lane == v) ? bal[v] : my;

        if (lane < 8) {                 // single divergent region, one b32 store
            myRow[jt] = my;
            degLo += __builtin_popcount(my & 0xffffu);
            degHi += __builtin_popcount(my >> 16);
        }
    }
    if (lane < 8) {
        deg[rt * 16 + lane]     = (int)degLo;
        deg[rt * 16 + lane + 8] = (int)degHi;
    }
}

// ---------------- kernel 2: core flags, comp init, per-tile core bitmasks -----------
__global__ __launch_bounds__(256) void core_kernel(const int* __restrict__ deg,
                                                   int* __restrict__ core,
                                                   int* __restrict__ comp,
                                                   unsigned short* __restrict__ corebits) {
    int i = blockIdx.x * blockDim.x + threadIdx.x;
    if (i >= NPTS) return;
    int c = (deg[i] >= MIN_PTS) ? 1 : 0;
    core[i] = c;
    comp[i] = c ? i : SENT;
    unsigned b = __builtin_amdgcn_ballot_w32(c != 0); // lanes align with 16-point tiles
    if ((threadIdx.x & 15) == 0) {
        unsigned half = ((threadIdx.x >> 4) & 1) ? (b >> 16) : (b & 0xffffu);
        corebits[i >> 4] = (unsigned short)half;
    }
}

// ---------------- kernel 3: adj &= core[column]  (adj becomes adj_core) -------------
// Both 16-bit halves of a packed word share the same 16 columns (tile = idx % NTILE).
__global__ __launch_bounds__(256) void maskadj_kernel(unsigned* __restrict__ adjB,
                                                      const unsigned short* __restrict__ corebits) {
    int idx = blockIdx.x * blockDim.x + threadIdx.x; // over (NPTS/2)*NTILE words
    if (idx >= (NPTS / 2) * NTILE) return;
    unsigned cb = (unsigned)corebits[idx & (NTILE - 1)];
    adjB[idx] &= cb * 0x00010001u;
}

// ---------------- wave-per-row neighbor min over adj_core ---------------------------
__device__ __forceinline__ int row_neighbor_min(const unsigned* __restrict__ rowWords,
                                                int shift, const int* __restrict__ comp,
                                                int lane) {
    int m = SENT;
#pragma unroll
    for (int rep = 0; rep < NTILE / 32; ++rep) {
        int t = lane + rep * 32;
        unsigned mask = (rowWords[t] >> shift) & 0xffffu;
        while (mask) {
            int bpos = __builtin_ctz(mask);
            mask &= mask - 1;
            m = imin(m, comp[t * 16 + bpos]);
        }
    }
#pragma unroll
    for (int off = 16; off > 0; off >>= 1)
        m = imin(m, __shfl_xor(m, off, 32));
    return m;
}

__device__ __forceinline__ const unsigned* row_ptr(const unsigned* adjB, int i, int* shift) {
    const int rt = i >> 4, r = i & 15;
    *shift = (r & 8) ? 16 : 0;
    return adjB + (size_t)(rt * 8 + (r & 7)) * NTILE;
}

// ---------------- kernel 4: async min-label propagation + pointer jumping -----------
__global__ __launch_bounds__(256) void prop_kernel(const unsigned* __restrict__ adjB,
                                                   const int* __restrict__ core,
                                                   int* __restrict__ comp) {
    const int wave = (blockIdx.x * blockDim.x + threadIdx.x) >> 5;
    const int lane = threadIdx.x & 31;
    if (wave >= NPTS) return;
    const int i = wave;
    int shift;
    const unsigned* row = row_ptr(adjB, i, &shift);
    int m = row_neighbor_min(row, shift, comp, lane);
    if (lane == 0 && core[i]) {
        int c = imin(comp[i], m);
        if (c < SENT) c = imin(c, comp[c]);   // pointer jump (stays in component, monotone)
        if (c < SENT) c = imin(c, comp[c]);
        comp[i] = c;                          // only this wave writes comp[i]; min is monotone
    }
}

// ---------------- kernel 5: point roots + is_root -----------------------------------
__global__ __launch_bounds__(256) void root_kernel(const unsigned* __restrict__ adjB,
                                                   const int* __restrict__ core,
                                                   const int* __restrict__ comp,
                                                   int* __restrict__ proot,
                                                   int* __restrict__ isroot) {
    const int wave = (blockIdx.x * blockDim.x + threadIdx.x) >> 5;
    const int lane = threadIdx.x & 31;
    if (wave >= NPTS) return;
    const int i = wave;
    int shift;
    const unsigned* row = row_ptr(adjB, i, &shift);
    int m = row_neighbor_min(row, shift, comp, lane); // nb_root
    if (lane == 0) {
        int c = comp[i];
        proot[i]  = core[i] ? c : m;
        isroot[i] = (core[i] && c == i) ? 1 : 0;
    }
}

// ---------------- kernel 6: compact cluster ids by (class, root index) --------------
__global__ __launch_bounds__(256) void rank_kernel(const int* __restrict__ isroot,
                                                   const int* __restrict__ cls,
                                                   int* __restrict__ rankArr) {
    int i = blockIdx.x * blockDim.x + threadIdx.x;
    if (i >= NPTS) return;
    if (!isroot[i]) { rankArr[i] = 0; return; }
    long long key = (long long)cls[i] * NPTS + i;
    int cnt = 0;
    for (int j = 0; j < NPTS; ++j) {
        if (isroot[j]) {
            long long kj = (long long)cls[j] * NPTS + j;
            cnt += (kj < key) ? 1 : 0;
        }
    }
    rankArr[i] = cnt;
}

// ---------------- kernel 7: emit (out, labels) --------------------------------------
__global__ __launch_bounds__(256) void out_kernel(const float* __restrict__ x,
                                                  const int* __restrict__ cls,
                                                  const int* __restrict__ proot,
                                                  const int* __restrict__ rankArr,
                                                  float* __restrict__ out) {
    int i = blockIdx.x * blockDim.x + threadIdx.x;
    if (i >= NPTS) return;
    int r = proot[i];
    bool clustered = r < SENT;
    int lab = clustered ? rankArr[imin(r, NPTS - 1)] : -1;
    const float* row = x + (size_t)i * NCOLS;
#pragma unroll
    for (int c = 0; c < 5; ++c)
        out[(size_t)i * OUTW + c] = clustered ? row[c] : 0.0f;
    out[(size_t)i * OUTW + 5] = clustered ? (float)cls[i] : 0.0f;
    out[(size_t)NPTS * OUTW + i] = (float)lab;   // labels, concatenated after `out`
}

// ---------------- launcher ----------------------------------------------------------
extern "C" void kernel_launch(void* const* d_in, const int* in_sizes, int n_in,
                              void* d_out, int out_size, void* d_ws, size_t ws_size,
                              hipStream_t stream) {
    const float* x = (const float*)d_in[0];
    float* out = (float*)d_out;

    // workspace carve-up (all 256B aligned)
    char* ws = (char*)d_ws;
    size_t off = 0;
    auto carve = [&](size_t bytes) -> char* {
        char* p = ws + off;
        off += (bytes + 255) & ~(size_t)255;
        return p;
    };
    unsigned*       adjB     = (unsigned*)carve((size_t)(NPTS / 2) * NTILE * 4); // 8 MB, L2-resident
    float*          coords4  = (float*)carve((size_t)NPTS * 4 * 4);              // 128 KB padded coords
    int*            deg      = (int*)carve(NPTS * 4);
    int*            cls      = (int*)carve(NPTS * 4);
    float*          sq       = (float*)carve(NPTS * 4);
    int*            comp     = (int*)carve(NPTS * 4);
    int*            core     = (int*)carve(NPTS * 4);
    int*            proot    = (int*)carve(NPTS * 4);
    int*            isroot   = (int*)carve(NPTS * 4);
    int*            rankArr  = (int*)carve(NPTS * 4);
    unsigned short* corebits = (unsigned short*)carve(NTILE * 2);

    prep_kernel<<<NPTS / 256, 256, 0, stream>>>(x, coords4, sq, cls);
    adj_kernel<<<NTILE, 32, 0, stream>>>(coords4, sq, cls, adjB, deg);
    core_kernel<<<NPTS / 256, 256, 0, stream>>>(deg, core, comp, corebits);
    maskadj_kernel<<<((NPTS / 2) * NTILE) / 256, 256, 0, stream>>>(adjB, corebits);

    // fixed iteration count (graph capture forbids device-dependent looping);
    // 40 rounds with pointer jumping vastly exceeds the eps-graph diameter.
    for (int it = 0; it < 40; ++it)
        prop_kernel<<<(NPTS * 32) / 256, 256, 0, stream>>>(adjB, core, comp);

    root_kernel<<<(NPTS * 32) / 256, 256, 0, stream>>>(adjB, core, comp, proot, isroot);
    rank_kernel<<<NPTS / 256, 256, 0, stream>>>(isroot, cls, rankArr);
    out_kernel<<<NPTS / 256, 256, 0, stream>>>(x, cls, proot, rankArr, out);
}